// InteractiveAttention_49649821942350
// MI455X (gfx1250) — compile-verified
//
#include <hip/hip_runtime.h>
#include <hip/hip_bf16.h>
#include <math.h>

// Problem constants (from reference setup_inputs)
#define BB 64
#define LC 512
#define LA 64
#define DC 1024
#define DA 1024
#define EPSV 1e-7f

typedef float v2f __attribute__((ext_vector_type(2)));
typedef float v8f __attribute__((ext_vector_type(8)));

// ---------------------------------------------------------------------------
// K1: asp_avg[b][d] = mean over l of asp_text[b][l][d]
// ---------------------------------------------------------------------------
__global__ void k_asp_mean(const float* __restrict__ asp, float* __restrict__ asp_avg) {
    int idx = blockIdx.x * blockDim.x + threadIdx.x;      // 0 .. B*DA-1
    int b = idx / DA;
    int d = idx % DA;
    const float* p = asp + (size_t)b * LA * DA + d;
    float s = 0.f;
    #pragma unroll 8
    for (int l = 0; l < LA; ++l) s += p[(size_t)l * DA];
    asp_avg[idx] = s * (1.0f / (float)LA);
}

// ---------------------------------------------------------------------------
// K2/K6: D[m][n] = sum_k A[m][k] * W[n][k]   (M=64, N=1024, K=1024)
// One wave (32 lanes) computes one 16x16 tile with V_WMMA_F32_16X16X4_F32.
// A layout (16x4 f32): lane = half*16 + m ; v2f = A[m][half*2 + {0,1}]
// B layout (4x16 f32): lane = half*16 + n ; v2f = B[half*2 + {0,1}][n] = W[n][half*2 + {0,1}]
// C/D layout: VGPR j holds row m0 + j + half*8, column n0 + (lane&15)
// ---------------------------------------------------------------------------
__global__ void k_wmma_gemm_nt(const float* __restrict__ A,   // [64 x 1024] row-major
                               const float* __restrict__ W,   // [1024 x 1024] row-major
                               float* __restrict__ D) {       // [64 x 1024] row-major
    const int lane = threadIdx.x;          // 0..31 (wave32)
    const int n0   = blockIdx.x * 16;
    const int m0   = blockIdx.y * 16;
    const int half = lane >> 4;            // 0 or 1
    const int lidx = lane & 15;
    const int K = 1024, N = 1024;

    const float* arow = A + (size_t)(m0 + lidx) * K + half * 2;
    const float* brow = W + (size_t)(n0 + lidx) * K + half * 2;

    v8f c = {};
    #pragma unroll 4
    for (int k = 0; k < K; k += 4) {
        v2f a = *(const v2f*)(arow + k);
        v2f b = *(const v2f*)(brow + k);
        c = __builtin_amdgcn_wmma_f32_16x16x4_f32(
                /*neg_a=*/false, a, /*neg_b=*/false, b,
                /*c_mod=*/(short)0, c, /*reuse_a=*/false, /*reuse_b=*/false);
    }

    float* drow = D + (size_t)(m0 + half * 8) * N + n0 + lidx;
    #pragma unroll
    for (int j = 0; j < 8; ++j) drow[(size_t)j * N] = c[j];
}

// ---------------------------------------------------------------------------
// K3: one streaming pass over context producing
//     s_c[b][l]          = dot(context[b,l,:], Vc[b,:])
//     partial[chunk,b,d] = sum over 64 rows of this chunk of context[b,l,d]
// Block: 256 threads = 8 waves, handles (b, chunk of 64 rows).
// Wave w handles rows chunk*64 + w*8 + r (r=0..7); lane covers 8 float4 segs.
// ---------------------------------------------------------------------------
__global__ void k_ctx_pass1(const float* __restrict__ ctx, const float* __restrict__ Vc,
                            float* __restrict__ s_c, float* __restrict__ partial) {
    const int chunk = blockIdx.x;          // 0..7
    const int b     = blockIdx.y;          // 0..63
    const int tid   = threadIdx.x;         // 0..255
    const int wave  = tid >> 5;
    const int lane  = tid & 31;

    const float* base = ctx + (size_t)b * LC * DC;
    const float* vb   = Vc + (size_t)b * DC;

    float4 v[8], acc[8];
    #pragma unroll
    for (int s = 0; s < 8; ++s) {
        v[s]   = *(const float4*)(vb + s * 128 + lane * 4);
        acc[s] = make_float4(0.f, 0.f, 0.f, 0.f);
    }

    for (int r = 0; r < 8; ++r) {
        const int l = chunk * 64 + wave * 8 + r;
        const float* row = base + (size_t)l * DC;
        float dot = 0.f;
        #pragma unroll
        for (int s = 0; s < 8; ++s) {
            float4 x = *(const float4*)(row + s * 128 + lane * 4);
            acc[s].x += x.x; acc[s].y += x.y; acc[s].z += x.z; acc[s].w += x.w;
            dot += x.x * v[s].x + x.y * v[s].y + x.z * v[s].z + x.w * v[s].w;
        }
        #pragma unroll
        for (int off = 16; off > 0; off >>= 1) dot += __shfl_xor(dot, off, 32);
        if (lane == 0) s_c[(size_t)b * LC + l] = dot;
    }

    // combine column sums across the 8 waves (deterministic, no atomics)
    __shared__ float4 sm4[8 * 256];                  // 32 KB LDS
    #pragma unroll
    for (int s = 0; s < 8; ++s) sm4[wave * 256 + s * 32 + lane] = acc[s];
    __syncthreads();

    float4 tot = make_float4(0.f, 0.f, 0.f, 0.f);
    #pragma unroll
    for (int w = 0; w < 8; ++w) {
        float4 p = sm4[w * 256 + tid];
        tot.x += p.x; tot.y += p.y; tot.z += p.z; tot.w += p.w;
    }
    *(float4*)(partial + ((size_t)chunk * BB + b) * DC + tid * 4) = tot;
}

// ---------------------------------------------------------------------------
// K4: ctx_avg[b][d] = (1/LC) * sum over 8 chunks of partial
// ---------------------------------------------------------------------------
__global__ void k_ctx_avg(const float* __restrict__ partial, float* __restrict__ cavg) {
    int idx = blockIdx.x * blockDim.x + threadIdx.x;      // 0 .. B*DC-1
    float s = 0.f;
    #pragma unroll
    for (int c = 0; c < 8; ++c) s += partial[(size_t)c * BB * DC + idx];
    cavg[idx] = s * (1.0f / (float)LC);
}

// ---------------------------------------------------------------------------
// K5: a_c[b][l] = exp(tanh(s_c + context_b)) normalized by (row-sum + EPS)
// One block of 256 threads per batch (each thread owns 2 of the 512 rows).
// ---------------------------------------------------------------------------
__global__ void k_ctx_attn(const float* __restrict__ s_c, const float* __restrict__ cb,
                           float* __restrict__ a_c) {
    const int b = blockIdx.x;
    const int tid = threadIdx.x;           // 0..255
    __shared__ float sm[256];
    __shared__ float inv;

    float e0 = __expf(tanhf(s_c[(size_t)b * LC + tid] + cb[tid]));
    float e1 = __expf(tanhf(s_c[(size_t)b * LC + tid + 256] + cb[tid + 256]));
    sm[tid] = e0 + e1;
    __syncthreads();
    for (int s = 128; s > 0; s >>= 1) {
        if (tid < s) sm[tid] += sm[tid + s];
        __syncthreads();
    }
    if (tid == 0) inv = 1.0f / (sm[0] + EPSV);
    __syncthreads();
    a_c[(size_t)b * LC + tid]       = e0 * inv;
    a_c[(size_t)b * LC + tid + 256] = e1 * inv;
}

// ---------------------------------------------------------------------------
// K7: attend_context[b][d] = sum_l context[b,l,d] * a_c[b,l]  -> out[:, 0:DC]
// Grid: (DC/256, B); a_c row staged in LDS; second (final) pass over context.
// ---------------------------------------------------------------------------
__global__ void k_attend_ctx(const float* __restrict__ ctx, const float* __restrict__ a_c,
                             float* __restrict__ out) {
    const int b   = blockIdx.y;
    const int d   = blockIdx.x * 256 + threadIdx.x;
    const int tid = threadIdx.x;

    __shared__ float ac[LC];
    ac[tid]       = a_c[(size_t)b * LC + tid];
    ac[tid + 256] = a_c[(size_t)b * LC + tid + 256];
    __syncthreads();

    const float* p = ctx + (size_t)b * LC * DC + d;
    float acc = 0.f;
    #pragma unroll 8
    for (int l = 0; l < LC; ++l) acc += p[(size_t)l * DC] * ac[l];
    out[(size_t)b * (DC + DA) + d] = acc;
}

// ---------------------------------------------------------------------------
// K8: aspect branch final (one block of 256 threads per batch):
//     s_t[b][l] = dot(asp_text[b,l,:], U[b,:]);  a_t = exp(tanh(s_t + aspect_b))
//     a_t_sum = sum_l a_t + EPS
//     attend_asp[b][d] = a_t_sum * LA * asp_avg[b][d]   (reference's replace-by-sum)
// ---------------------------------------------------------------------------
__global__ void k_aspect_final(const float* __restrict__ asp, const float* __restrict__ U,
                               const float* __restrict__ ab, const float* __restrict__ asp_avg,
                               float* __restrict__ out) {
    const int b    = blockIdx.x;
    const int tid  = threadIdx.x;          // 0..255
    const int wave = tid >> 5;
    const int lane = tid & 31;

    __shared__ float wsum[8];
    __shared__ float scale;

    const float* base = asp + (size_t)b * LA * DA;
    const float* ub   = U + (size_t)b * DA;

    float4 u[8];
    #pragma unroll
    for (int s = 0; s < 8; ++s) u[s] = *(const float4*)(ub + s * 128 + lane * 4);

    float asum = 0.f;
    for (int r = 0; r < 8; ++r) {
        const int l = wave * 8 + r;
        const float* row = base + (size_t)l * DA;
        float dot = 0.f;
        #pragma unroll
        for (int s = 0; s < 8; ++s) {
            float4 x = *(const float4*)(row + s * 128 + lane * 4);
            dot += x.x * u[s].x + x.y * u[s].y + x.z * u[s].z + x.w * u[s].w;
        }
        #pragma unroll
        for (int off = 16; off > 0; off >>= 1) dot += __shfl_xor(dot, off, 32);
        if (lane == 0) asum += __expf(tanhf(dot + ab[l]));
    }
    if (lane == 0) wsum[wave] = asum;
    __syncthreads();
    if (tid == 0) {
        float t = 0.f;
        #pragma unroll
        for (int w = 0; w < 8; ++w) t += wsum[w];
        scale = (t + EPSV) * (float)LA;    // a_t_sum * La  (asp_avg is the mean)
    }
    __syncthreads();

    float4 av = *(const float4*)(asp_avg + (size_t)b * DA + tid * 4);
    float4 o;
    o.x = av.x * scale; o.y = av.y * scale; o.z = av.z * scale; o.w = av.w * scale;
    *(float4*)(out + (size_t)b * (DC + DA) + DC + tid * 4) = o;
}

// ---------------------------------------------------------------------------
extern "C" void kernel_launch(void* const* d_in, const int* in_sizes, int n_in,
                              void* d_out, int out_size, void* d_ws, size_t ws_size,
                              hipStream_t stream) {
    (void)in_sizes; (void)n_in; (void)out_size; (void)ws_size;

    const float* context   = (const float*)d_in[0];   // [B, LC, DC]
    const float* asp_text  = (const float*)d_in[1];   // [B, LA, DA]
    const float* context_w = (const float*)d_in[2];   // [DC, DA]
    const float* context_b = (const float*)d_in[3];   // [LC]
    const float* aspect_w  = (const float*)d_in[4];   // [DA, DC]
    const float* aspect_b  = (const float*)d_in[5];   // [LA]
    float* out = (float*)d_out;                       // [B, DC+DA]

    // workspace layout (floats)
    float* ws       = (float*)d_ws;
    float* asp_avg  = ws;                    // B*DA   = 65536
    float* Vc       = asp_avg + BB * DA;     // B*DC   = 65536
    float* U        = Vc + BB * DC;          // B*DA   = 65536
    float* s_c      = U + BB * DA;           // B*LC   = 32768
    float* a_c      = s_c + BB * LC;         // B*LC   = 32768
    float* cavg     = a_c + BB * LC;         // B*DC   = 65536
    float* partial  = cavg + BB * DC;        // 8*B*DC = 524288   (~3.3 MB total)

    // 1) asp_avg = mean_l(asp_text)
    k_asp_mean<<<dim3((BB * DA) / 256), dim3(256), 0, stream>>>(asp_text, asp_avg);

    // 2) Vc[b,:] = asp_avg[b,:] x context_w^T   (WMMA f32)
    k_wmma_gemm_nt<<<dim3(DC / 16, BB / 16), dim3(32), 0, stream>>>(asp_avg, context_w, Vc);

    // 3) fused context pass: s_c + per-chunk column partial sums
    k_ctx_pass1<<<dim3(LC / 64, BB), dim3(256), 0, stream>>>(context, Vc, s_c, partial);

    // 4) ctx_avg from partials
    k_ctx_avg<<<dim3((BB * DC) / 256), dim3(256), 0, stream>>>(partial, cavg);

    // 5) a_c = normalize(exp(tanh(s_c + context_b)))
    k_ctx_attn<<<dim3(BB), dim3(256), 0, stream>>>(s_c, context_b, a_c);

    // 6) U[b,:] = cavg[b,:] x aspect_w^T   (WMMA f32)
    k_wmma_gemm_nt<<<dim3(DA / 16, BB / 16), dim3(32), 0, stream>>>(cavg, aspect_w, U);

    // 7) attend_context -> out[:, 0:DC]   (second/final pass over context)
    k_attend_ctx<<<dim3(DC / 256, BB), dim3(256), 0, stream>>>(context, a_c, out);

    // 8) aspect branch final -> out[:, DC:DC+DA]
    k_aspect_final<<<dim3(BB), dim3(256), 0, stream>>>(asp_text, U, aspect_b, asp_avg, out);
}